// HGNNConv_19327352832290
// MI455X (gfx1250) — compile-verified
//
#include <hip/hip_runtime.h>

typedef float v2f __attribute__((ext_vector_type(2)));
typedef float v8f __attribute__((ext_vector_type(8)));

#define D        128
#define LDS_PAD  132   // 128 + 4 floats: 16 lanes hit distinct bank pairs

// ---------------------------------------------------------------------------
// Zero he (E*128 in ws, E read from device scalar) and d_out (N*128).
// ---------------------------------------------------------------------------
__global__ void __launch_bounds__(256)
zero_kernel(float4* __restrict__ he4, float4* __restrict__ out4,
            const int* __restrict__ nE_ptr, long long nOut4)
{
    long long e4     = (long long)(*nE_ptr) * (D / 4);
    long long stride = (long long)gridDim.x * blockDim.x;
    long long i0     = (long long)blockIdx.x * blockDim.x + threadIdx.x;
    float4 z = make_float4(0.f, 0.f, 0.f, 0.f);
    for (long long i = i0; i < e4;    i += stride) he4[i]  = z;
    for (long long i = i0; i < nOut4; i += stride) out4[i] = z;
}

// ---------------------------------------------------------------------------
// Phase 1: he[cols[i]] += vals[i] * embeds[rows[i]]   (one wave per nnz)
// ---------------------------------------------------------------------------
__global__ void __launch_bounds__(256)
scatter_edges_kernel(const float* __restrict__ embeds, const float* __restrict__ vals,
                     const int* __restrict__ rows, const int* __restrict__ cols,
                     float* __restrict__ he, int nnz)
{
    int w = (int)((blockIdx.x * (unsigned)blockDim.x + threadIdx.x) >> 5);
    if (w >= nnz) return;
    int lane = threadIdx.x & 31;
    int   r = rows[w];
    int   c = cols[w];
    float v = vals[w];
    const float4 x = *(const float4*)(embeds + (size_t)r * D + lane * 4);
    float* dst = he + (size_t)c * D + lane * 4;
    atomicAdd(dst + 0, v * x.x);
    atomicAdd(dst + 1, v * x.y);
    atomicAdd(dst + 2, v * x.z);
    atomicAdd(dst + 3, v * x.w);
}

// ---------------------------------------------------------------------------
// Phase 2: lat1 = LayerNorm(he @ W^T) via V_WMMA_F32_16X16X4_F32.
// Block = 256 threads = 8 waves; wave w owns output columns [16w, 16w+16).
// Each wave keeps its B fragments (16 W columns) in 64 VGPRs across all
// E/16 row-stripes (grid-stride).  lat1 lives at ws + E*128.
// ---------------------------------------------------------------------------
__global__ void __launch_bounds__(256)
gemm_ln_kernel(float* __restrict__ ws, const float* __restrict__ Wm,
               const float* __restrict__ g1, const float* __restrict__ b1,
               const int* __restrict__ nE_ptr)
{
    const int E = *nE_ptr;
    const float* he  = ws;
    float*      lat1 = ws + (size_t)E * D;

    __shared__ float ldsT[16 * LDS_PAD];

    const int tid  = threadIdx.x;
    const int wave = tid >> 5;
    const int lane = tid & 31;
    const int lid  = lane & 15;
    const int half = lane >> 4;

    // --- preload B fragments: B[k][n] = W[n][k]; frag kk = W[n][4kk+2h .. +1]
    const int n_global = wave * 16 + lid;                 // 0..127
    const float* wrow  = Wm + (size_t)n_global * D;
    v2f breg[32];
#pragma unroll
    for (int kk = 0; kk < 32; ++kk)
        breg[kk] = *(const v2f*)(wrow + kk * 4 + 2 * half);

    // --- preload LN affine params for this thread's 8 columns
    const int rRow = tid >> 4;        // 0..15 : row handled in LN phase
    const int sub  = tid & 15;        // 16 threads cooperate per row
    float gg[8], bb[8];
#pragma unroll
    for (int j = 0; j < 8; ++j) { gg[j] = g1[sub * 8 + j]; bb[j] = b1[sub * 8 + j]; }

    const int nStripes = (E + 15) >> 4;
    for (int s = blockIdx.x; s < nStripes; s += gridDim.x) {
        const int e0 = s << 4;

        // cooperative load: 16 rows x 128 floats of he into LDS (padded rows)
        {
            const int er = e0 + rRow;
            const int c8 = sub * 8;
            float4 x0 = make_float4(0.f, 0.f, 0.f, 0.f), x1 = x0;
            if (er < E) {
                const float4* src = (const float4*)(he + (size_t)er * D + c8);
                x0 = src[0]; x1 = src[1];
            }
            float* dst = &ldsT[rRow * LDS_PAD + c8];
            *(float4*)(dst)     = x0;
            *(float4*)(dst + 4) = x1;
        }
        __syncthreads();

        // 32 x V_WMMA_F32_16X16X4_F32 : full f32 GEMM, K=128
        v8f acc = {0.f, 0.f, 0.f, 0.f, 0.f, 0.f, 0.f, 0.f};
#pragma unroll
        for (int kk = 0; kk < 32; ++kk) {
            v2f a = *(const v2f*)&ldsT[lid * LDS_PAD + kk * 4 + 2 * half];
            acc = __builtin_amdgcn_wmma_f32_16x16x4_f32(
                      false, a, false, breg[kk], (short)0, acc, false, false);
        }
        __syncthreads();   // everyone done reading ldsT; reuse it for C tile

        // scatter C tile: VGPR v, lane (half,lid) -> row v+8*half, col n_global
#pragma unroll
        for (int v = 0; v < 8; ++v)
            ldsT[(v + 8 * half) * LDS_PAD + n_global] = acc[v];
        __syncthreads();

        // fused LayerNorm: 16 threads per row, 8 elements each
        {
            const float* rowp = &ldsT[rRow * LDS_PAD + sub * 8];
            float x[8], s1 = 0.f, s2 = 0.f;
#pragma unroll
            for (int j = 0; j < 8; ++j) { x[j] = rowp[j]; s1 += x[j]; s2 += x[j] * x[j]; }
#pragma unroll
            for (int m = 1; m < 16; m <<= 1) {
                s1 += __shfl_xor(s1, m, 32);
                s2 += __shfl_xor(s2, m, 32);
            }
            float mean = s1 * (1.f / 128.f);
            float var  = s2 * (1.f / 128.f) - mean * mean;
            float rstd = rsqrtf(var + 1e-5f);
            const int er = e0 + rRow;
            if (er < E) {
                float* outp = lat1 + (size_t)er * D + sub * 8;
#pragma unroll
                for (int j = 0; j < 8; ++j)
                    outp[j] = (x[j] - mean) * rstd * gg[j] + bb[j];
            }
        }
        __syncthreads();   // before next stripe overwrites ldsT
    }
}

// ---------------------------------------------------------------------------
// Phase 3: out[rows[i]] += vals[i] * lat1[cols[i]]   (one wave per nnz)
// ---------------------------------------------------------------------------
__global__ void __launch_bounds__(256)
scatter_nodes_kernel(const float* __restrict__ ws, const float* __restrict__ vals,
                     const int* __restrict__ rows, const int* __restrict__ cols,
                     float* __restrict__ out, const int* __restrict__ nE_ptr, int nnz)
{
    int w = (int)((blockIdx.x * (unsigned)blockDim.x + threadIdx.x) >> 5);
    if (w >= nnz) return;
    const float* lat1 = ws + (size_t)(*nE_ptr) * D;
    int lane = threadIdx.x & 31;
    int   r = rows[w];
    int   c = cols[w];
    float v = vals[w];
    const float4 x = *(const float4*)(lat1 + (size_t)c * D + lane * 4);
    float* dst = out + (size_t)r * D + lane * 4;
    atomicAdd(dst + 0, v * x.x);
    atomicAdd(dst + 1, v * x.y);
    atomicAdd(dst + 2, v * x.z);
    atomicAdd(dst + 3, v * x.w);
}

// ---------------------------------------------------------------------------
// Phase 4: out = leaky_relu(LayerNorm(out)) in place (one wave per node row)
// ---------------------------------------------------------------------------
__global__ void __launch_bounds__(256)
ln_lrelu_kernel(float* __restrict__ out, const float* __restrict__ g2,
                const float* __restrict__ b2, int N)
{
    int row = (int)((blockIdx.x * (unsigned)blockDim.x + threadIdx.x) >> 5);
    if (row >= N) return;
    int lane = threadIdx.x & 31;
    float* p = out + (size_t)row * D + lane * 4;
    float4 x = *(const float4*)p;
    float s1 = x.x + x.y + x.z + x.w;
    float s2 = x.x * x.x + x.y * x.y + x.z * x.z + x.w * x.w;
#pragma unroll
    for (int m = 1; m < 32; m <<= 1) {
        s1 += __shfl_xor(s1, m, 32);
        s2 += __shfl_xor(s2, m, 32);
    }
    float mean = s1 * (1.f / 128.f);
    float var  = s2 * (1.f / 128.f) - mean * mean;
    float rstd = rsqrtf(var + 1e-5f);
    float4 g = *(const float4*)(g2 + lane * 4);
    float4 b = *(const float4*)(b2 + lane * 4);
    float4 y;
    y.x = (x.x - mean) * rstd * g.x + b.x;
    y.y = (x.y - mean) * rstd * g.y + b.y;
    y.z = (x.z - mean) * rstd * g.z + b.z;
    y.w = (x.w - mean) * rstd * g.w + b.w;
    y.x = y.x > 0.f ? y.x : 0.2f * y.x;
    y.y = y.y > 0.f ? y.y : 0.2f * y.y;
    y.z = y.z > 0.f ? y.z : 0.2f * y.z;
    y.w = y.w > 0.f ? y.w : 0.2f * y.w;
    *(float4*)p = y;
}

// ---------------------------------------------------------------------------
extern "C" void kernel_launch(void* const* d_in, const int* in_sizes, int n_in,
                              void* d_out, int out_size, void* d_ws, size_t ws_size,
                              hipStream_t stream)
{
    const float* embeds = (const float*)d_in[0];
    const float* vals   = (const float*)d_in[1];
    const float* Wm     = (const float*)d_in[2];
    const float* g1     = (const float*)d_in[3];
    const float* b1     = (const float*)d_in[4];
    const float* g2     = (const float*)d_in[5];
    const float* b2     = (const float*)d_in[6];
    const int*   rows   = (const int*)d_in[7];
    const int*   cols   = (const int*)d_in[8];
    const int*   nE     = (const int*)d_in[10];   // device scalar: n_edges

    float* out = (float*)d_out;
    float* ws  = (float*)d_ws;   // [0, E*128) = he ; [E*128, 2*E*128) = lat1

    const int N   = in_sizes[0] / D;
    const int NNZ = in_sizes[1];

    zero_kernel<<<2048, 256, 0, stream>>>((float4*)ws, (float4*)out, nE,
                                          (long long)N * (D / 4));

    const int sblocks = (NNZ + 7) / 8;     // 8 waves (one nnz each) per block
    scatter_edges_kernel<<<sblocks, 256, 0, stream>>>(embeds, vals, rows, cols, ws, NNZ);

    gemm_ln_kernel<<<2048, 256, 0, stream>>>(ws, Wm, g1, b1, nE);

    scatter_nodes_kernel<<<sblocks, 256, 0, stream>>>(ws, vals, rows, cols, out, nE, NNZ);

    ln_lrelu_kernel<<<(N + 7) / 8, 256, 0, stream>>>(out, g2, b2, N);
}